// AssociativeEmbeddingLoss_88656714925445
// MI455X (gfx1250) — compile-verified
//
#include <hip/hip_runtime.h>
#include <stdint.h>

// Associative-embedding loss for MI455X (gfx1250, wave32).
//
// Shape: N=32, K=17, H=W=256, M=30. Only 16,320 scalars are gathered from the
// 142MB tag map -> the kernel is latency bound, so we use exactly N waves
// (one per image, one lane per person) and do all cross-person work with
// wave32 shuffles (ds_bpermute) plus the matrix unit:
//   exp(-(ai-aj)^2) = exp(-(ai^2 + aj^2 - 2*ai*aj))
// The rank-1 outer product P = mean * mean^T (32x32) is computed as four
// 16x16 tiles on V_WMMA_F32_16X16X4_F32 with the vector in K-row 0.

namespace {

constexpr int kN = 32, kK = 17, kH = 256, kW = 256, kM = 30;
constexpr int kKHW = kK * kH * kW;

typedef float v2f __attribute__((ext_vector_type(2)));
typedef float v8f __attribute__((ext_vector_type(8)));

__device__ __forceinline__ float wave_sum32(float x) {
#pragma unroll
  for (int off = 16; off > 0; off >>= 1) x += __shfl_xor(x, off, 32);
  return x;
}

__global__ __launch_bounds__(32) void ae_per_batch(
    const float* __restrict__ tags,
    const long long* __restrict__ joints,   // int64 [N, M, K, 2]
    float* __restrict__ ws) {               // ws[0:N]=pull_i, ws[N:2N]=push_i
  const int n = blockIdx.x;
  const int lane = threadIdx.x;             // lane == person id (lanes 30,31 idle)
  const float* __restrict__ tflat = tags + (size_t)n * kKHW;

  // ---- per-person gather: keep the K gathered tags in registers (2-pass pull)
  float gv[kK];
  unsigned vmask = 0u;
  float cnt = 0.f, sum_g = 0.f;
  if (lane < kM) {
    const long long* __restrict__ jp = joints + (((size_t)n * kM + lane) * kK) * 2;
#pragma unroll
    for (int k = 0; k < kK; ++k) {
      const long long loc = jp[2 * k];
      const bool vis = jp[2 * k + 1] > 0;
      float g = 0.f;
      if (vis) {
        g = tflat[loc];                     // random gather, in-range by construction
        vmask |= (1u << k);
        cnt += 1.f;
        sum_g += g;
      }
      gv[k] = g;
    }
  } else {
#pragma unroll
    for (int k = 0; k < kK; ++k) gv[k] = 0.f;
  }

  const float safe_cnt = fmaxf(cnt, 1.f);
  const float mean = sum_g / safe_cnt;      // 0 for invalid persons (sum_g==0)
  const float vf = (cnt > 0.f) ? 1.f : 0.f; // person_valid

  // ---- pull: exact two-pass variance like the reference
  float ss = 0.f;
#pragma unroll
  for (int k = 0; k < kK; ++k) {
    const float w = ((vmask >> k) & 1u) ? 1.f : 0.f;
    const float d = gv[k] - mean;
    ss += w * d * d;
  }
  const float pull_pp = ss / safe_cnt;      // already 0 when person invalid
  const float pull_n = wave_sum32(pull_pp * vf);
  const float num_tags = wave_sum32(vf);

  // ---- push: P = mean (x) mean via V_WMMA_F32_16X16X4_F32, 4 tiles of 16x16.
  // A (16x4 f32) layout per ISA: VGPR0 lanes 0-15 hold (M=lane, K=0); all other
  // K slots zeroed. B (4x16) mirrors with N across lanes -> the two operands
  // for the low/high halves of the vector are identical for A and B.
  const float mlo = __shfl(mean, lane & 15, 32);        // a[lane&15]
  const float mhi = __shfl(mean, 16 + (lane & 15), 32); // a[16 + (lane&15)]
  const float keep = (lane < 16) ? 1.f : 0.f;           // zero K=2 row (lanes 16-31)
  v2f Vlo; Vlo[0] = mlo * keep; Vlo[1] = 0.f;           // VGPR1 = K=1/K=3 rows = 0
  v2f Vhi; Vhi[0] = mhi * keep; Vhi[1] = 0.f;
  const v8f zero = {};

  v8f P[2][2];
  P[0][0] = __builtin_amdgcn_wmma_f32_16x16x4_f32(false, Vlo, false, Vlo, (short)0, zero, false, false);
  P[0][1] = __builtin_amdgcn_wmma_f32_16x16x4_f32(false, Vlo, false, Vhi, (short)0, zero, false, false);
  P[1][0] = __builtin_amdgcn_wmma_f32_16x16x4_f32(false, Vhi, false, Vlo, (short)0, zero, false, false);
  P[1][1] = __builtin_amdgcn_wmma_f32_16x16x4_f32(false, Vhi, false, Vhi, (short)0, zero, false, false);

  // Decode accumulators (C/D 16x16 f32: VGPR r, lanes 0-15 -> M=r, N=lane;
  // lanes 16-31 -> M=r+8) and apply exp(-(sq_r + sq_c - 2*P)) with masks.
  const float sq = mean * mean;
  const int lcol = lane & 15;
  const int lhalf = (lane >> 4) << 3;       // 0 or 8
  float acc = 0.f;
#pragma unroll
  for (int ti = 0; ti < 2; ++ti) {
#pragma unroll
    for (int tj = 0; tj < 2; ++tj) {
      const int col = tj * 16 + lcol;
      const float sq_c = __shfl(sq, col, 32);
      const float v_c = __shfl(vf, col, 32);
#pragma unroll
      for (int r = 0; r < 8; ++r) {
        const int row = ti * 16 + lhalf + r;
        const float sq_r = __shfl(sq, row, 32);
        const float v_r = __shfl(vf, row, 32);
        const float d2 = sq_r + sq_c - 2.f * P[ti][tj][r];
        const float e = expf(-d2);
        acc += (row == col) ? 0.f : v_r * v_c * e;   // off-diagonal, both valid
      }
    }
  }
  const float push_sum = wave_sum32(acc);

  if (lane == 0) {
    const float n_pairs = num_tags * (num_tags - 1.f);
    ws[n] = pull_n / fmaxf(num_tags, 1.f);
    ws[kN + n] = (n_pairs > 0.f) ? (0.5f * push_sum / fmaxf(n_pairs, 1.f)) : 0.f;
  }
}

// Final mean over the N=32 per-image losses: exactly one wave32.
__global__ __launch_bounds__(32) void ae_reduce(const float* __restrict__ ws,
                                                float* __restrict__ out) {
  const int lane = threadIdx.x;
  float p = ws[lane];
  float q = ws[kN + lane];
  p = wave_sum32(p);
  q = wave_sum32(q);
  if (lane == 0) {
    out[0] = p * (1.f / kN);   // pull_loss
    out[1] = q * (1.f / kN);   // push_loss
  }
}

}  // namespace

extern "C" void kernel_launch(void* const* d_in, const int* in_sizes, int n_in,
                              void* d_out, int out_size, void* d_ws, size_t ws_size,
                              hipStream_t stream) {
  const float* tags = (const float*)d_in[0];            // float32 [N,K,H,W]
  const long long* joints = (const long long*)d_in[1];  // int64   [N,M,K,2]
  float* ws = (float*)d_ws;                             // needs 2*N floats (256 B)
  float* out = (float*)d_out;                           // {pull_loss, push_loss}

  ae_per_batch<<<kN, 32, 0, stream>>>(tags, joints, ws);
  ae_reduce<<<1, 32, 0, stream>>>(ws, out);
}